// ReEig_5781025980865
// MI455X (gfx1250) — compile-verified
//
#include <hip/hip_runtime.h>

// ReEig via matrix-sign Newton-Schulz, all compute on V_WMMA_F32_16X16X4_F32.
// One workgroup (4 x wave32) per 64x64 matrix.
//   ReEig(P) = 0.5*(P + eps*I + |P - eps*I|),  |A| = sign(A)*A
//   sign(A): over-scaled cubic NS  Y <- c1*Y - c3*Y^3  (c1=1.5s, c3=0.5s^3)
//   s=1.45 for 12 "lift" iterations (growth x2.175/iter; max of f on [0,1.45]
//   is f(1)=1 and 1.45 < sqrt(3), so spectrum stays in [0,1]), then s=1 polish.
// All B operands (Y, T=Y^2, A) are symmetric, so B fragments are read
// transposed: contiguous in an LDS row -> single ds_load_b64 per fragment,
// no VGPR re-pairing movs.  T = Y*Y^T is then bitwise symmetric.

#define MATN        64
#define LSTR        68            // padded LDS row stride (floats); 68*4B = 17*16B
#define AGG_ITERS   12
#define POL_ITERS   5
#define NS_ITERS    (AGG_ITERS + POL_ITERS)
#define EPS_RE      1e-4f
#define C1_AGG      2.175f        // 1.5 * 1.45
#define C3_AGG      1.5243125f    // 0.5 * 1.45^3

typedef float v2f __attribute__((ext_vector_type(2)));
typedef float v4f __attribute__((ext_vector_type(4)));
typedef float v8f __attribute__((ext_vector_type(8)));

// This wave's A-fragments (its 16 rows, all K) into registers, v2f (b64) loads.
// ISA 7.12.2 f32 16x4 A layout: lane = M; VGPR0/1 = K {0,1} (lanes 0-15) or
// K {2,3} (lanes 16-31) of each 4-wide K step.
__device__ __forceinline__ void load_afrag(const float* As, v2f af[16],
                                           int rbase, int half, int idx)
{
    const float* row = As + (rbase + idx) * LSTR + 2 * half;   // 8B aligned
#pragma unroll
    for (int k = 0; k < 16; ++k)
        af[k] = *(const v2f*)&row[4 * k];
}

// 16x64 row block (4 tiles) of A * Bs, Bs SYMMETRIC (fragments read transposed).
__device__ __forceinline__ void gemm_rowblock_symB(const v2f af[16], const float* Bs,
                                                   v8f acc[4], int half, int idx)
{
#pragma unroll
    for (int c = 0; c < 4; ++c)
        acc[c] = v8f{0.f, 0.f, 0.f, 0.f, 0.f, 0.f, 0.f, 0.f};

#pragma unroll   // full unroll: af[] indexed by constants only
    for (int k = 0; k < 16; ++k) {
        const int kb = 4 * k;
#pragma unroll
        for (int c = 0; c < 4; ++c) {
            // Need b = {B[kb+2h][col], B[kb+2h+1][col]}; B symmetric ->
            // read row `col` contiguously: one ds_load_b64, even VGPR pair.
            const v2f b = *(const v2f*)&Bs[(16 * c + idx) * LSTR + kb + 2 * half];
            acc[c] = __builtin_amdgcn_wmma_f32_16x16x4_f32(
                false, af[k], false, b, (short)0, acc[c], false, false);
        }
    }
}

__global__ __launch_bounds__(128)
void reeig_ns_kernel(const float* __restrict__ P, float* __restrict__ out)
{
    __shared__ __attribute__((aligned(16))) float Amat[MATN * LSTR]; // A = P - eps*I
    __shared__ __attribute__((aligned(16))) float Ymat[MATN * LSTR]; // sign iterate
    __shared__ __attribute__((aligned(16))) float Tmat[MATN * LSTR]; // Y^2 scratch
    __shared__ float red[4];

    const int tid   = threadIdx.x;
    const int wave  = tid >> 5;
    const int lane  = tid & 31;
    const int half  = lane >> 4;
    const int idx   = lane & 15;
    const int rbase = wave * 16;
    const size_t base = (size_t)blockIdx.x * MATN * MATN;

    // ---- load A = P - eps*I (b128 loads), accumulate ||A||_F^2 ----
    float ss = 0.f;
    const v4f* Pv = (const v4f*)(P + base);
    for (int i = tid; i < MATN * MATN / 4; i += 128) {
        v4f v = Pv[i];
        const int r = (4 * i) >> 6, c = (4 * i) & 63;
#pragma unroll
        for (int j = 0; j < 4; ++j) if (c + j == r) v[j] -= EPS_RE;
        ss += v[0] * v[0] + v[1] * v[1] + v[2] * v[2] + v[3] * v[3];
        *(v4f*)&Amat[r * LSTR + c] = v;
    }
    for (int off = 16; off > 0; off >>= 1) ss += __shfl_down(ss, off, 32);
    if (lane == 0) red[wave] = ss;
    __syncthreads();
    const float fro2 = red[0] + red[1] + red[2] + red[3];
    const float inv  = rsqrtf(fmaxf(fro2, 1e-30f)); // ||A||_F >= ||A||_2: spectrum in [0,1]

    // ---- Y0 = A / ||A||_F ----
    for (int i = tid; i < MATN * MATN / 4; i += 128) {
        const int r = (4 * i) >> 6, c = (4 * i) & 63;
        v4f v = *(const v4f*)&Amat[r * LSTR + c];
        *(v4f*)&Ymat[r * LSTR + c] = v * inv;
    }

    v8f acc[4];
    v2f af[16];

    // ---- Newton-Schulz sign iteration (scaled lift, then polish) ----
#pragma unroll 1
    for (int it = 0; it < NS_ITERS; ++it) {
        const float c1 = (it < AGG_ITERS) ? C1_AGG : 1.5f;
        const float c3 = (it < AGG_ITERS) ? C3_AGG : 0.5f;
        __syncthreads();                               // Y stable for all waves
        load_afrag(Ymat, af, rbase, half, idx);        // own rows, reused twice
        gemm_rowblock_symB(af, Ymat, acc, half, idx);  // T-rows = Y*Y^T (bitwise sym)
#pragma unroll
        for (int c = 0; c < 4; ++c)
#pragma unroll
            for (int r = 0; r < 8; ++r)
                Tmat[(rbase + r + 8 * half) * LSTR + 16 * c + idx] = acc[c][r];
        __syncthreads();                               // T visible to all waves
        gemm_rowblock_symB(af, Tmat, acc, half, idx);  // G-rows = Y*T
        // Update only this wave's own rows (others read T, not Y, here).
#pragma unroll
        for (int c = 0; c < 4; ++c)
#pragma unroll
            for (int r = 0; r < 8; ++r) {
                const int row = rbase + r + 8 * half, col = 16 * c + idx;
                const float y = Ymat[row * LSTR + col];
                Ymat[row * LSTR + col] = c1 * y - c3 * acc[c][r];
            }
    }

    // ---- |A| = sign(A)*A = Y*A ;  out = 0.5*A + eps*I + 0.5*|A| ----
    __syncthreads();
    load_afrag(Ymat, af, rbase, half, idx);
    gemm_rowblock_symB(af, Amat, acc, half, idx);      // A symmetric
#pragma unroll
    for (int c = 0; c < 4; ++c)
#pragma unroll
        for (int r = 0; r < 8; ++r) {
            const int row = rbase + r + 8 * half, col = 16 * c + idx;
            float o = 0.5f * Amat[row * LSTR + col] + 0.5f * acc[c][r];
            if (row == col) o += EPS_RE;
            out[base + row * MATN + col] = o;
        }
}

extern "C" void kernel_launch(void* const* d_in, const int* in_sizes, int n_in,
                              void* d_out, int out_size, void* d_ws, size_t ws_size,
                              hipStream_t stream)
{
    const float* P = (const float*)d_in[0];
    float* out     = (float*)d_out;
    const int batch = in_sizes[0] / (MATN * MATN);   // 2048
    reeig_ns_kernel<<<batch, 128, 0, stream>>>(P, out);
}